// TreeCrossAttention_31550829756811
// MI455X (gfx1250) — compile-verified
//
#include <hip/hip_runtime.h>

typedef __attribute__((ext_vector_type(2))) float v2f;
typedef __attribute__((ext_vector_type(4))) float v4f;
typedef __attribute__((ext_vector_type(8))) float v8f;

static __device__ __forceinline__ v8f wmma_f32x4(v2f a, v2f b, v8f c) {
  // D = A(16x4,f32) x B(4x16,f32) + C(16x16,f32)
  return __builtin_amdgcn_wmma_f32_16x16x4_f32(false, a, false, b, (short)0, c, false, false);
}

// ---------------------------------------------------------------------------
// Phase 1: spiking tree over key_value -> vadd[128][32]  (== vadd[B=16][D=256])
// Grid: 8 blocks x 256 threads (8 wave32). Block g owns bh-rows [16g, 16g+16).
// WMMA mapping: M = 16 bh-rows (one tile), N = 32 outputs (2 tiles), K = 32 (8x4).
// ---------------------------------------------------------------------------
__global__ __launch_bounds__(256) void tca_tree_kernel(
    const float* __restrict__ key_value,  // [16][16][256]
    const float* __restrict__ W,          // [15][32][32]  W[p][o][d]
    const float* __restrict__ bias,       // [15][32]
    const float* __restrict__ thr,        // [15][32]
    const float* __restrict__ tau,        // [15][32]
    const float* __restrict__ vres,       // [15][32]
    float* __restrict__ vadd)             // [128][32]
{
  __shared__ float KA[8][16][32];
  __shared__ float VA[8][16][32];
  __shared__ float KB[4][16][32];
  __shared__ float VB[4][16][32];

  const int tid  = threadIdx.x;
  const int wave = tid >> 5;
  const int lane = tid & 31;
  const int base = blockIdx.x * 16;   // first bh-row of this workgroup
  const int ml   = lane & 15;         // M (for A) or N (for B/C/D) index
  const int hi   = lane >> 4;         // half-wave selector

  // ---- Leaf level: nodes l=0..7 use W[7+l]; wave == node; xk == xv ----
  {
    const int l = wave;
    const int p = 7 + l;
    const float* Wp = W + p * 1024;
    const int r_a = base + ml;
    const float* xrow = key_value + (((r_a >> 3) * 16 + l) * 256) + (r_a & 7) * 32;
    for (int t = 0; t < 2; ++t) {
      v8f acc = {};
      const int o = t * 16 + ml;
      for (int k0 = 0; k0 < 8; ++k0) {
        const int kk = k0 * 4 + hi * 2;
        v2f a;  a.x  = xrow[kk];          a.y  = xrow[kk + 1];
        v2f bm; bm.x = Wp[o * 32 + kk];   bm.y = Wp[o * 32 + kk + 1];
        acc = wmma_f32x4(a, bm, acc);
      }
      const float bl = bias[p * 32 + o];
      const float th = thr [p * 32 + o];
      const float ta = tau [p * 32 + o];
      const float vr = vres[p * 32 + o];
      for (int j = 0; j < 8; ++j) {
        const int m = j + hi * 8;
        const int r = base + m;
        const float x  = key_value[(((r >> 3) * 16 + l) * 256) + (r & 7) * 32 + o];
        const float ok = acc[j] + bl;
        const bool  sk = (x >= th);
        const float va = sk ? vr : x;
        const float v2 = ta * va + x;
        KA[l][m][o] = sk ? ok : 0.0f;
        VA[l][m][o] = (v2 >= th) ? ok : 0.0f;
      }
    }
  }
  __syncthreads();

  // ---- Level n=4: nodes l=0..3 use W[3+l]; wave -> (node, path) ----
  {
    const int l    = wave >> 1;
    const int path = wave & 1;          // 0 = k-path, 1 = v-path
    const int p    = 3 + l;
    const float* Wp = W + p * 1024;
    for (int t = 0; t < 2; ++t) {
      v8f acc = {};
      const int o = t * 16 + ml;
      for (int k0 = 0; k0 < 8; ++k0) {
        const int kk = k0 * 4 + hi * 2;
        v2f a;
        if (path == 0) {
          a.x = 0.5f * (KA[2*l][ml][kk]     + KA[2*l+1][ml][kk]);
          a.y = 0.5f * (KA[2*l][ml][kk + 1] + KA[2*l+1][ml][kk + 1]);
        } else {
          a.x = 0.5f * (VA[2*l][ml][kk]     + VA[2*l+1][ml][kk]);
          a.y = 0.5f * (VA[2*l][ml][kk + 1] + VA[2*l+1][ml][kk + 1]);
        }
        v2f bm; bm.x = Wp[o * 32 + kk]; bm.y = Wp[o * 32 + kk + 1];
        acc = wmma_f32x4(a, bm, acc);
      }
      const float bl = bias[p * 32 + o];
      const float th = thr [p * 32 + o];
      const float ta = tau [p * 32 + o];
      const float vr = vres[p * 32 + o];
      for (int j = 0; j < 8; ++j) {
        const int m  = j + hi * 8;
        const float fk = 0.5f * (KA[2*l][m][o] + KA[2*l+1][m][o]);
        const float ok = acc[j] + bl;
        const bool  sk = (fk >= th);
        if (path == 0) {
          KB[l][m][o] = sk ? ok : 0.0f;
        } else {
          const float fv = 0.5f * (VA[2*l][m][o] + VA[2*l+1][m][o]);
          const float va = sk ? vr : fk;
          const float v2 = ta * va + fv;
          VB[l][m][o] = (v2 >= th) ? ok : 0.0f;
        }
      }
    }
  }
  __syncthreads();

  // ---- Level n=2: nodes l=0..1 use W[1+l]; wave -> (node, path, tile) ----
  {
    const int l    = (wave >> 2) & 1;
    const int path = (wave >> 1) & 1;
    const int t    = wave & 1;
    const int p    = 1 + l;
    const float* Wp = W + p * 1024;
    v8f acc = {};
    const int o = t * 16 + ml;
    for (int k0 = 0; k0 < 8; ++k0) {
      const int kk = k0 * 4 + hi * 2;
      v2f a;
      if (path == 0) {
        a.x = 0.5f * (KB[2*l][ml][kk]     + KB[2*l+1][ml][kk]);
        a.y = 0.5f * (KB[2*l][ml][kk + 1] + KB[2*l+1][ml][kk + 1]);
      } else {
        a.x = 0.5f * (VB[2*l][ml][kk]     + VB[2*l+1][ml][kk]);
        a.y = 0.5f * (VB[2*l][ml][kk + 1] + VB[2*l+1][ml][kk + 1]);
      }
      v2f bm; bm.x = Wp[o * 32 + kk]; bm.y = Wp[o * 32 + kk + 1];
      acc = wmma_f32x4(a, bm, acc);
    }
    const float bl = bias[p * 32 + o];
    const float th = thr [p * 32 + o];
    const float ta = tau [p * 32 + o];
    const float vr = vres[p * 32 + o];
    for (int j = 0; j < 8; ++j) {
      const int m  = j + hi * 8;
      const float fk = 0.5f * (KB[2*l][m][o] + KB[2*l+1][m][o]);
      const float ok = acc[j] + bl;
      const bool  sk = (fk >= th);
      if (path == 0) {
        KA[l][m][o] = sk ? ok : 0.0f;
      } else {
        const float fv = 0.5f * (VB[2*l][m][o] + VB[2*l+1][m][o]);
        const float va = sk ? vr : fk;
        const float v2 = ta * va + fv;
        VA[l][m][o] = (v2 >= th) ? ok : 0.0f;
      }
    }
  }
  __syncthreads();

  // ---- Root: node 0, W[0]; softmax over 1 element == 1, only v-path matters ----
  if (wave < 2) {
    const int t = wave;
    v8f acc = {};
    const int o = t * 16 + ml;
    for (int k0 = 0; k0 < 8; ++k0) {
      const int kk = k0 * 4 + hi * 2;
      v2f a;
      a.x = 0.5f * (VA[0][ml][kk]     + VA[1][ml][kk]);
      a.y = 0.5f * (VA[0][ml][kk + 1] + VA[1][ml][kk + 1]);
      v2f bm; bm.x = W[o * 32 + kk]; bm.y = W[o * 32 + kk + 1];
      acc = wmma_f32x4(a, bm, acc);
    }
    const float bl = bias[o];
    const float th = thr [o];
    const float ta = tau [o];
    const float vr = vres[o];
    for (int j = 0; j < 8; ++j) {
      const int m = j + hi * 8;
      const int r = base + m;
      const float fk = 0.5f * (KA[0][m][o] + KA[1][m][o]);
      const float fv = 0.5f * (VA[0][m][o] + VA[1][m][o]);
      const float ov = acc[j] + bl;
      const bool  sk = (fk >= th);
      const float va = sk ? vr : fk;
      const float v2 = ta * va + fv;
      vadd[r * 32 + o] = (v2 >= th) ? ov : 0.0f;   // r*32+o == b*256 + (h*32+o)
    }
  }
}

// ---------------------------------------------------------------------------
// Phase 2: out = LayerNorm(query + vadd[b])  — pure HBM bandwidth (~256 MB).
// One wave32 per 256-float row; b128 non-temporal loads/stores; shfl reduce.
// Grid: (16*8192)/8 blocks x 256 threads (8 rows per block, all same b).
// ---------------------------------------------------------------------------
__global__ __launch_bounds__(256) void tca_fuse_kernel(
    const float* __restrict__ query,  // [16][8192][256]
    const float* __restrict__ vadd,   // [16][256]
    const float* __restrict__ gamma,  // [256]
    const float* __restrict__ beta,   // [256]
    float* __restrict__ out)          // [16][8192][256]
{
  __shared__ float sV[256];
  __shared__ float sG[256];
  __shared__ float sB[256];
  const int tid = threadIdx.x;
  const long long row0 = (long long)blockIdx.x * 8;
  const int b = (int)(row0 >> 13);    // 8192 rows per batch; 8 | 8192
  sV[tid] = vadd[b * 256 + tid];
  sG[tid] = gamma[tid];
  sB[tid] = beta[tid];
  __syncthreads();

  const int wave = tid >> 5;
  const int lane = tid & 31;
  const long long row = row0 + wave;
  const float* q = query + row * 256;
  float*       o = out   + row * 256;
  const int c0 = lane * 8;

  const v4f q0 = __builtin_nontemporal_load((const v4f*)(q + c0));
  const v4f q1 = __builtin_nontemporal_load((const v4f*)(q + c0 + 4));
  float x[8];
  x[0] = q0.x + sV[c0 + 0]; x[1] = q0.y + sV[c0 + 1];
  x[2] = q0.z + sV[c0 + 2]; x[3] = q0.w + sV[c0 + 3];
  x[4] = q1.x + sV[c0 + 4]; x[5] = q1.y + sV[c0 + 5];
  x[6] = q1.z + sV[c0 + 6]; x[7] = q1.w + sV[c0 + 7];

  float s = 0.0f, s2 = 0.0f;
#pragma unroll
  for (int i = 0; i < 8; ++i) { s += x[i]; s2 += x[i] * x[i]; }
#pragma unroll
  for (int off = 16; off > 0; off >>= 1) {
    s  += __shfl_xor(s,  off, 32);
    s2 += __shfl_xor(s2, off, 32);
  }
  const float mu   = s * (1.0f / 256.0f);
  const float var  = s2 * (1.0f / 256.0f) - mu * mu;
  const float rstd = rsqrtf(var + 1e-5f);

  v4f r0, r1;
  r0.x = sG[c0 + 0] * (x[0] - mu) * rstd + sB[c0 + 0];
  r0.y = sG[c0 + 1] * (x[1] - mu) * rstd + sB[c0 + 1];
  r0.z = sG[c0 + 2] * (x[2] - mu) * rstd + sB[c0 + 2];
  r0.w = sG[c0 + 3] * (x[3] - mu) * rstd + sB[c0 + 3];
  r1.x = sG[c0 + 4] * (x[4] - mu) * rstd + sB[c0 + 4];
  r1.y = sG[c0 + 5] * (x[5] - mu) * rstd + sB[c0 + 5];
  r1.z = sG[c0 + 6] * (x[6] - mu) * rstd + sB[c0 + 6];
  r1.w = sG[c0 + 7] * (x[7] - mu) * rstd + sB[c0 + 7];
  __builtin_nontemporal_store(r0, (v4f*)(o + c0));
  __builtin_nontemporal_store(r1, (v4f*)(o + c0 + 4));
}

extern "C" void kernel_launch(void* const* d_in, const int* in_sizes, int n_in,
                              void* d_out, int out_size, void* d_ws, size_t ws_size,
                              hipStream_t stream) {
  const float* query = (const float*)d_in[0];
  const float* kv    = (const float*)d_in[1];
  const float* W     = (const float*)d_in[2];
  const float* b     = (const float*)d_in[3];
  const float* thr   = (const float*)d_in[4];
  const float* tau   = (const float*)d_in[5];
  const float* vres  = (const float*)d_in[6];
  const float* gamma = (const float*)d_in[7];
  const float* beta  = (const float*)d_in[8];
  float* out  = (float*)d_out;
  float* vadd = (float*)d_ws;             // 128*32 floats = 16 KB scratch

  tca_tree_kernel<<<8, 256, 0, stream>>>(kv, W, b, thr, tau, vres, vadd);

  const int rows   = 16 * 8192;           // B * SQ
  const int blocks = rows / 8;            // 8 rows (waves) per block
  tca_fuse_kernel<<<blocks, 256, 0, stream>>>(query, vadd, gamma, beta, out);
}